// HungarianMatcher_43284680409696
// MI455X (gfx1250) — compile-verified
//
#include <hip/hip_runtime.h>

// Hungarian matcher cost matrix for MI455X (gfx1250, wave32).
//
// C[i,j] = 5*L1(pred_pt[i], tgt_pt[j]) + 2*mean_c BCE(x[i,c], t[j,c])
//        = 5*L1 + [softplus(x[i,0]) + softplus(x[i,1])] - sum_c x[i,c]*t[j,c]
//
// The bilinear term is computed with V_WMMA_F32_16X16X4_F32 (K=2 padded to 4).
// Output (230 MB) is streamed with non-temporal stores; inputs staged in LDS.

typedef __attribute__((ext_vector_type(2))) float v2f;
typedef __attribute__((ext_vector_type(8))) float v8f;

#define N_PRED 12000   // 8 * 1500
#define N_TGT  4800
#define COST_CLASS_W 2.0f
#define COST_POINT_W 5.0f

__device__ __forceinline__ float softplusf(float x) {
  // log(1 + exp(x)), numerically stable
  return fmaxf(x, 0.0f) + log1pf(__expf(-fabsf(x)));
}

__global__ __launch_bounds__(256) void hungarian_cost_kernel(
    const float* __restrict__ pred_logits,   // [12000, 2]
    const float* __restrict__ pred_points,   // [12000, 2]
    const float* __restrict__ tgt_labels,    // [4800, 2]
    const float* __restrict__ tgt_points,    // [4800, 2]
    float* __restrict__ out)                 // [12000, 4800]
{
  // Block tile: 32 preds x 64 tgts. 8 waves, each does one 16x16 WMMA tile.
  __shared__ float s_t0[64], s_t1[64], s_tx[64], s_ty[64];
  __shared__ float s_x0[32], s_x1[32], s_px[32], s_py[32], s_bias[32];

  const int tid    = threadIdx.x;
  const int j_base = blockIdx.x * 64;
  const int i_base = blockIdx.y * 32;

  if (tid < 64) {                       // stage target tile
    const int j = j_base + tid;
    s_t0[tid] = tgt_labels[2 * j + 0];
    s_t1[tid] = tgt_labels[2 * j + 1];
    s_tx[tid] = tgt_points[2 * j + 0];
    s_ty[tid] = tgt_points[2 * j + 1];
  } else if (tid < 96) {                // stage prediction tile + per-row bias
    const int m = tid - 64;
    const int i = i_base + m;
    const float x0 = pred_logits[2 * i + 0];
    const float x1 = pred_logits[2 * i + 1];
    s_x0[m] = x0;
    s_x1[m] = x1;
    s_px[m] = pred_points[2 * i + 0];
    s_py[m] = pred_points[2 * i + 1];
    s_bias[m] = softplusf(x0) + softplusf(x1);  // = 2*class const term
  }
  __syncthreads();

  const int wave    = tid >> 5;
  const int lane    = tid & 31;
  const int p       = wave >> 2;        // pred sub-tile 0..1
  const int q       = wave & 3;         // tgt  sub-tile 0..3
  const int mlo     = p * 16;
  const int nlo     = q * 16;
  const int lane_hi = lane >> 4;        // 0 or 1
  const int lane_lo = lane & 15;

  // A fragment (16x4 f32): lanes 0-15 -> (M=lane, K=0..1 in v0/v1),
  // lanes 16-31 -> K=2..3 (zero pad). B fragment (4x16) mirrored on N.
  v2f a, b;
  a.x = (lane < 16) ? s_x0[mlo + lane_lo] : 0.0f;
  a.y = (lane < 16) ? s_x1[mlo + lane_lo] : 0.0f;
  b.x = (lane < 16) ? s_t0[nlo + lane_lo] : 0.0f;
  b.y = (lane < 16) ? s_t1[nlo + lane_lo] : 0.0f;

  v8f c = {};
  // S[m,n] = sum_c x[i0+m,c] * t[j0+n,c]
  c = __builtin_amdgcn_wmma_f32_16x16x4_f32(
      /*neg_a=*/false, a, /*neg_b=*/false, b,
      /*c_mod=*/(short)0, c, /*reuse_a=*/false, /*reuse_b=*/false);

  // C/D layout: VGPR r, lanes 0-15 -> M=r, lanes 16-31 -> M=r+8; N = lane&15.
  const float tx = s_tx[nlo + lane_lo];
  const float ty = s_ty[nlo + lane_lo];
  const int   j  = j_base + nlo + lane_lo;

#pragma unroll
  for (int r = 0; r < 8; ++r) {
    const int   ml = mlo + r + lane_hi * 8;
    const int   i  = i_base + ml;
    const float l1 = fabsf(s_px[ml] - tx) + fabsf(s_py[ml] - ty);
    const float val = COST_POINT_W * l1 + s_bias[ml] - c[r];
    __builtin_nontemporal_store(val, out + (size_t)i * N_TGT + j);
  }
}

extern "C" void kernel_launch(void* const* d_in, const int* in_sizes, int n_in,
                              void* d_out, int out_size, void* d_ws, size_t ws_size,
                              hipStream_t stream) {
  const float* pred_logits = (const float*)d_in[0];
  const float* pred_points = (const float*)d_in[1];
  const float* tgt_labels  = (const float*)d_in[2];
  const float* tgt_points  = (const float*)d_in[3];
  float* out = (float*)d_out;

  dim3 grid(N_TGT / 64, N_PRED / 32);   // 75 x 375 blocks
  hungarian_cost_kernel<<<grid, dim3(256), 0, stream>>>(
      pred_logits, pred_points, tgt_labels, tgt_points, out);
}